// CustomSAGE_68092411511561
// MI455X (gfx1250) — compile-verified
//
#include <hip/hip_runtime.h>
#include <hip/hip_bf16.h>

#define N_NODES   100000
#define N_EDGES   1600000
#define FDIM      128
#define N_CLASSES 32
#define N_GRAPHS  512

#define BM    64          // rows per block in the WMMA GEMM
#define LDSP  132         // padded LDS row stride (floats): 132 % 64 = 4 -> conflict-free

typedef __attribute__((ext_vector_type(2))) float v2f;
typedef __attribute__((ext_vector_type(8))) float v8f;
typedef __attribute__((ext_vector_type(4))) int   v4i;

#define AS3 __attribute__((address_space(3)))

#if defined(__has_builtin)
#  if __has_builtin(__builtin_amdgcn_global_load_async_to_lds_b128) && \
      __has_builtin(__builtin_amdgcn_s_wait_asynccnt)
#    define USE_ASYNC_COPY 1
#  endif
#endif
#ifndef USE_ASYNC_COPY
#  define USE_ASYNC_COPY 0
#endif

// ---------------- degree / histogram kernels ----------------

__global__ __launch_bounds__(256)
void deg_kernel(const long long* __restrict__ dst, float* __restrict__ deg, int n_edges) {
    int e = blockIdx.x * 256 + threadIdx.x;
    if (e < n_edges) atomicAdd(&deg[(int)dst[e]], 1.0f);
}

__global__ __launch_bounds__(256)
void inv_kernel(float* __restrict__ deg, int n) {
    int i = blockIdx.x * 256 + threadIdx.x;
    if (i < n) deg[i] = 1.0f / fmaxf(deg[i], 1.0f);
}

__global__ __launch_bounds__(256)
void gcnt_kernel(const long long* __restrict__ batch, float* __restrict__ gcnt, int n_nodes) {
    int i = blockIdx.x * 256 + threadIdx.x;
    if (i < n_nodes) atomicAdd(&gcnt[(int)batch[i]], 1.0f);
}

// ---------------- edge scatter-add: one wave per edge ----------------
// agg[dst] += h[src]; lane handles 4 contiguous floats (float4 gather, 4 atomic f32 adds).
// Working set (51.2 MB features) is fully L2-resident; this is an L2 atomic-throughput phase.

__global__ __launch_bounds__(256)
void scatter_add_kernel(const float* __restrict__ h,
                        const long long* __restrict__ src,
                        const long long* __restrict__ dst,
                        float* __restrict__ agg, int n_edges) {
    long long gid = (long long)blockIdx.x * 256 + threadIdx.x;
    int e = (int)(gid >> 5);
    if (e >= n_edges) return;
    int c4 = ((int)gid & 31) << 2;                 // 0..124
    int s = (int)src[e];
    int d = (int)dst[e];
    const float4 v = *(const float4*)(h + (size_t)s * FDIM + c4);
    float* p = agg + (size_t)d * FDIM + c4;
    atomicAdd(p + 0, v.x);
    atomicAdd(p + 1, v.y);
    atomicAdd(p + 2, v.z);
    atomicAdd(p + 3, v.w);
}

// ---------------- fused SAGE layer GEMM (fp32 WMMA) ----------------
// hout = relu_opt( (agg .* invdeg) @ Wl^T + bl + hin @ Wr^T )
// 256 threads = 8 wave32. Wave w owns output cols [16w,16w+16).
// 4 accumulators (one per 16-row sub-tile) so each weight tile is loaded once per k-step.
// Interior blocks (all rows valid) take fully unguarded staging/store paths.

template<int RELU>
__global__ __launch_bounds__(256)
void sage_gemm_kernel(const float* __restrict__ agg, const float* __restrict__ invdeg,
                      const float* __restrict__ hin,
                      const float* __restrict__ Wl, const float* __restrict__ bl,
                      const float* __restrict__ Wr,
                      float* __restrict__ hout, int n_nodes) {
    __shared__ float sA[BM * LDSP];   // raw aggregated neighbor features
    __shared__ float sX[BM * LDSP];   // self features

    const int t = threadIdx.x;
    const int row0 = blockIdx.x * BM;
    const bool edge_block = (row0 + BM > n_nodes);   // uniform across block

    if (edge_block) {
        // zero-fill so out-of-range rows feed clean zeros into the WMMAs
        for (int i = t; i < BM * LDSP; i += 256) { sA[i] = 0.f; sX[i] = 0.f; }
        __syncthreads();
    }

#if USE_ASYNC_COPY
    // CDNA5 async DMA: global -> LDS without a VGPR round-trip (ASYNCcnt-tracked).
    if (!edge_block) {
        #pragma unroll
        for (int i = 0; i < 4; ++i) {
            int idx = t + i * 256;                  // float4 slot 0..1023
            int r   = idx >> 5;                     // 0..31
            int c4  = (idx & 31) << 2;              // 0..124
            int row = row0 + r;
            __builtin_amdgcn_global_load_async_to_lds_b128(
                (v4i*)const_cast<float*>(agg + (size_t)row * FDIM + c4),
                (AS3 v4i*)(sA + r * LDSP + c4), 0, 0);
            __builtin_amdgcn_global_load_async_to_lds_b128(
                (v4i*)const_cast<float*>(hin + (size_t)row * FDIM + c4),
                (AS3 v4i*)(sX + r * LDSP + c4), 0, 0);
        }
    } else {
        #pragma unroll
        for (int i = 0; i < 4; ++i) {
            int idx = t + i * 256;
            int r   = idx >> 5;
            int c4  = (idx & 31) << 2;
            int row = row0 + r;
            if (row < n_nodes) {
                __builtin_amdgcn_global_load_async_to_lds_b128(
                    (v4i*)const_cast<float*>(agg + (size_t)row * FDIM + c4),
                    (AS3 v4i*)(sA + r * LDSP + c4), 0, 0);
                __builtin_amdgcn_global_load_async_to_lds_b128(
                    (v4i*)const_cast<float*>(hin + (size_t)row * FDIM + c4),
                    (AS3 v4i*)(sX + r * LDSP + c4), 0, 0);
            }
        }
    }
    __builtin_amdgcn_s_wait_asynccnt(0);
#else
    #pragma unroll
    for (int i = 0; i < 8; ++i) {
        int idx = t + i * 256;                      // 0..2047
        int r   = idx >> 5;                         // 0..63
        int c4  = (idx & 31) << 2;                  // 0..124
        int row = row0 + r;
        if (row < n_nodes) {
            *(float4*)(sA + r * LDSP + c4) = *(const float4*)(agg + (size_t)row * FDIM + c4);
            *(float4*)(sX + r * LDSP + c4) = *(const float4*)(hin + (size_t)row * FDIM + c4);
        }
    }
#endif
    __syncthreads();

    const int wave  = t >> 5;                 // 0..7 (wave32)
    const int lane  = t & 31;
    const int lm    = lane & 15;
    const int nl    = wave * 16 + lm;         // output column handled by this lane
    const int khalf = (lane >> 4) * 2;        // A/B layout: lanes 0-15 -> K 0/1, 16-31 -> K 2/3
    const float bias = bl[nl];

    // per-sub-tile mean reciprocal: this lane's A row is fixed (M = lm) per sub-tile
    float invA[4];
    #pragma unroll
    for (int rt = 0; rt < 4; ++rt) {
        int row = row0 + rt * 16 + lm;
        invA[rt] = (row < n_nodes) ? invdeg[row] : 0.0f;
    }

    v8f acc[4];
    #pragma unroll
    for (int rt = 0; rt < 4; ++rt) {
        #pragma unroll
        for (int r = 0; r < 8; ++r) acc[rt][r] = bias;
    }

    #pragma unroll 2
    for (int k0 = 0; k0 < FDIM; k0 += 4) {
        const int kk = k0 + khalf;
        // weight tiles loaded once per k-step, reused across 4 row sub-tiles
        v2f bw_l = *(const v2f*)(Wl + (size_t)nl * FDIM + kk);
        v2f bw_r = *(const v2f*)(Wr + (size_t)nl * FDIM + kk);
        #pragma unroll
        for (int rt = 0; rt < 4; ++rt) {
            const int arow = rt * 16 + lm;
            v2f a0 = *(const v2f*)(sA + arow * LDSP + kk);
            a0 = a0 * invA[rt];   // fused scatter-mean divide
            acc[rt] = __builtin_amdgcn_wmma_f32_16x16x4_f32(
                          false, a0, false, bw_l, (short)0, acc[rt], false, false);
            v2f a1 = *(const v2f*)(sX + arow * LDSP + kk);
            acc[rt] = __builtin_amdgcn_wmma_f32_16x16x4_f32(
                          false, a1, false, bw_r, (short)0, acc[rt], false, false);
        }
    }

    // C/D layout: VGPR r holds M = rt*16 + r + (lane>=16 ? 8 : 0), N = nl
    const int rhalf = (lane >> 4) * 8;
    if (!edge_block) {
        #pragma unroll
        for (int rt = 0; rt < 4; ++rt) {
            float* base = hout + (size_t)(row0 + rt * 16 + rhalf) * FDIM + nl;
            #pragma unroll
            for (int r = 0; r < 8; ++r) {
                float v = acc[rt][r];
                if (RELU) v = fmaxf(v, 0.0f);
                base[(size_t)r * FDIM] = v;     // folds into immediate store offsets
            }
        }
    } else {
        #pragma unroll
        for (int rt = 0; rt < 4; ++rt) {
            #pragma unroll
            for (int r = 0; r < 8; ++r) {
                int row = row0 + rt * 16 + rhalf + r;
                if (row < n_nodes) {
                    float v = acc[rt][r];
                    if (RELU) v = fmaxf(v, 0.0f);
                    hout[(size_t)row * FDIM + nl] = v;
                }
            }
        }
    }
}

// ---------------- global mean pool (scatter-add) ----------------

__global__ __launch_bounds__(256)
void pool_kernel(const float* __restrict__ h, const long long* __restrict__ batch,
                 float* __restrict__ gsum, int n_nodes) {
    long long gid = (long long)blockIdx.x * 256 + threadIdx.x;
    int node = (int)(gid >> 5);
    if (node >= n_nodes) return;
    int c4 = ((int)gid & 31) << 2;
    int b = (int)batch[node];
    const float4 v = *(const float4*)(h + (size_t)node * FDIM + c4);
    float* p = gsum + (size_t)b * FDIM + c4;
    atomicAdd(p + 0, v.x);
    atomicAdd(p + 1, v.y);
    atomicAdd(p + 2, v.z);
    atomicAdd(p + 3, v.w);
}

// ---------------- final tiny linear: out = (gsum/cnt) @ Wlin^T + blin ----------------

__global__ __launch_bounds__(256)
void final_kernel(const float* __restrict__ gsum, const float* __restrict__ gcnt,
                  const float* __restrict__ Wlin, const float* __restrict__ blin,
                  float* __restrict__ out) {
    int id = blockIdx.x * 256 + threadIdx.x;
    if (id >= N_GRAPHS * N_CLASSES) return;
    int g = id >> 5;              // N_CLASSES == 32
    int c = id & 31;
    float inv = 1.0f / fmaxf(gcnt[g], 1.0f);
    float acc = blin[c];
    #pragma unroll 8
    for (int k = 0; k < FDIM; ++k)
        acc = fmaf(gsum[(size_t)g * FDIM + k] * inv, Wlin[(size_t)c * FDIM + k], acc);
    out[(size_t)g * N_CLASSES + c] = acc;
}

// ---------------- host orchestration ----------------

extern "C" void kernel_launch(void* const* d_in, const int* in_sizes, int n_in,
                              void* d_out, int out_size, void* d_ws, size_t ws_size,
                              hipStream_t stream) {
    const float*     x     = (const float*)d_in[0];
    const long long* eidx  = (const long long*)d_in[1];
    const long long* batch = (const long long*)d_in[2];
    const float* Wl[3] = { (const float*)d_in[3], (const float*)d_in[6], (const float*)d_in[9]  };
    const float* bl[3] = { (const float*)d_in[4], (const float*)d_in[7], (const float*)d_in[10] };
    const float* Wr[3] = { (const float*)d_in[5], (const float*)d_in[8], (const float*)d_in[11] };
    const float* Wlin  = (const float*)d_in[12];
    const float* blin  = (const float*)d_in[13];

    const long long* src = eidx;
    const long long* dst = eidx + N_EDGES;

    const size_t featBytes = (size_t)N_NODES * FDIM * sizeof(float);
    char* ws = (char*)d_ws;
    float* agg  = (float*)ws;  ws += featBytes;
    float* h1   = (float*)ws;  ws += featBytes;
    float* h2   = (float*)ws;  ws += featBytes;
    float* deg  = (float*)ws;  ws += (size_t)N_NODES * sizeof(float);
    float* gsum = (float*)ws;  ws += (size_t)N_GRAPHS * FDIM * sizeof(float);
    float* gcnt = (float*)ws;  ws += (size_t)N_GRAPHS * sizeof(float);

    const int edgeBlocks  = ((long long)N_EDGES * 32 + 255) / 256;   // wave per edge
    const int nodeBlocks  = (N_NODES + 255) / 256;
    const int gemmBlocks  = (N_NODES + BM - 1) / BM;

    // in-degree -> reciprocal (computed fresh every call; deterministic)
    (void)hipMemsetAsync(deg, 0, (size_t)N_NODES * sizeof(float), stream);
    deg_kernel<<<(N_EDGES + 255) / 256, 256, 0, stream>>>(dst, deg, N_EDGES);
    inv_kernel<<<nodeBlocks, 256, 0, stream>>>(deg, N_NODES);

    const float* hin = x;
    float* houts[3] = { h1, h2, h1 };
    for (int l = 0; l < 3; ++l) {
        (void)hipMemsetAsync(agg, 0, featBytes, stream);
        scatter_add_kernel<<<edgeBlocks, 256, 0, stream>>>(hin, src, dst, agg, N_EDGES);
        if (l < 2) {
            sage_gemm_kernel<1><<<gemmBlocks, 256, 0, stream>>>(agg, deg, hin,
                                                                Wl[l], bl[l], Wr[l],
                                                                houts[l], N_NODES);
        } else {
            sage_gemm_kernel<0><<<gemmBlocks, 256, 0, stream>>>(agg, deg, hin,
                                                                Wl[l], bl[l], Wr[l],
                                                                houts[l], N_NODES);
        }
        hin = houts[l];
    }

    // global mean pool + classifier head
    (void)hipMemsetAsync(gcnt, 0, (size_t)N_GRAPHS * sizeof(float), stream);
    (void)hipMemsetAsync(gsum, 0, (size_t)N_GRAPHS * FDIM * sizeof(float), stream);
    gcnt_kernel<<<nodeBlocks, 256, 0, stream>>>(batch, gcnt, N_NODES);
    pool_kernel<<<((long long)N_NODES * 32 + 255) / 256, 256, 0, stream>>>(hin, batch, gsum, N_NODES);
    final_kernel<<<(N_GRAPHS * N_CLASSES + 255) / 256, 256, 0, stream>>>(gsum, gcnt, Wlin, blin,
                                                                         (float*)d_out);
}